// BINN_73237782331418
// MI455X (gfx1250) — compile-verified
//
#include <hip/hip_runtime.h>

// BINN forward for MI455X (gfx1250, wave32).
//  - Sparse gather layers: VALU FMAs; src/w indexed by blockIdx.y -> wave-uniform
//    scalar loads; lanes cover batch (coalesced-ish rows, x is L2-resident: 32MB << 192MB L2).
//  - LayerNorm+tanh: one block per batch row, two-pass reduction (faithful to reference).
//  - Dense heads (a @ hW, OUT=2): V_WMMA_F32_16X16X4_F32, one wave per 16-row tile.
//  - Final sparse layer (src4/w4) is dead code in the reference -> skipped.

#define IN_DIM   8000
#define H1_DIM   4000
#define H2_DIM   2000
#define H3_DIM   1000
#define OUT_DIM  2
#define FAN      16
#define BATCH    1024
#define LN_EPS   1e-5f

typedef __attribute__((ext_vector_type(2))) float v2f;
typedef __attribute__((ext_vector_type(8))) float v8f;

// z[b,n] = bias[n] + sum_f aprev[b, src[n,f]] * w[n,f]
// grid = (BATCH/256, H), block = 256. n uniform per block -> s_load for src/w.
__global__ void binn_sparse(const float* __restrict__ aprev,
                            const int*   __restrict__ src,
                            const float* __restrict__ w,
                            const float* __restrict__ bias,
                            float*       __restrict__ z,
                            int prev, int H)
{
    const int n = blockIdx.y;
    const int b = blockIdx.x * blockDim.x + threadIdx.x;
    const float* row = aprev + (size_t)b * prev;
    const int*   sp  = src + n * FAN;
    const float* wp  = w   + n * FAN;
    float acc = bias[n];
#pragma unroll
    for (int f = 0; f < FAN; ++f) {
        acc += row[sp[f]] * wp[f];
    }
    z[(size_t)b * H + n] = acc;
}

// In-place: z[row, :] = tanh( (z - mean)/sqrt(var + eps) * g + b ), per batch row.
__global__ void binn_ln_tanh(float* __restrict__ z,
                             const float* __restrict__ gv,
                             const float* __restrict__ bv,
                             int H)
{
    const int row = blockIdx.x;
    const int t   = threadIdx.x;
    float* zp = z + (size_t)row * H;
    __shared__ float red[256];

    // pass 1: mean
    float s = 0.0f;
    for (int i = t; i < H; i += 256) s += zp[i];
    red[t] = s; __syncthreads();
    for (int o = 128; o > 0; o >>= 1) {
        if (t < o) red[t] += red[t + o];
        __syncthreads();
    }
    const float m = red[0] / (float)H;
    __syncthreads();   // WAR: everyone read red[0] before it is overwritten

    // pass 2: variance = mean((z - m)^2)   (matches reference form)
    float s2 = 0.0f;
    for (int i = t; i < H; i += 256) { float d = zp[i] - m; s2 += d * d; }
    red[t] = s2; __syncthreads();
    for (int o = 128; o > 0; o >>= 1) {
        if (t < o) red[t] += red[t + o];
        __syncthreads();
    }
    const float r = rsqrtf(red[0] / (float)H + LN_EPS);

    for (int i = t; i < H; i += 256) {
        zp[i] = tanhf((zp[i] - m) * r * gv[i] + bv[i]);
    }
}

// p[tile*16 .. tile*16+15, 0..OUT-1] = a_tile @ hW + hb using V_WMMA_F32_16X16X4_F32.
// One wave (32 threads) per block; grid = BATCH/16.
// A frag (16x4 f32): lane L holds M = L&15; low lanes K=0,1 in {v0,v1}; high lanes K=2,3.
// B frag (4x16 f32): lane L holds N = L&15; low lanes rows K=0,1; high lanes rows K=2,3.
// D (16x16 f32): c[r] on lane L is element (M = r + (L<16?0:8), N = L&15).
__global__ void binn_head_wmma(const float* __restrict__ a,
                               const float* __restrict__ hW,   // [H, OUT] row-major
                               const float* __restrict__ hb,   // [OUT]
                               float*       __restrict__ p,    // [B, OUT]
                               int H)
{
    const int lane  = threadIdx.x;        // 0..31, full wave (EXEC all-1s at WMMA)
    const int tile  = blockIdx.x;         // 16 batch rows
    const int m     = lane & 15;
    const int khalf = (lane >> 4) * 2;    // 0 for lanes 0-15, 2 for lanes 16-31
    const bool valid = (m < OUT_DIM);
    const float* arow = a + (size_t)(tile * 16 + m) * H + khalf;
    const float* wcol = hW + (valid ? m : 0);         // safe pointer for all lanes
    const float  msk  = valid ? 1.0f : 0.0f;          // zero out N>=OUT columns

    v8f c = {};
    for (int k = 0; k < H; k += 4) {
        v2f af = *(const v2f*)(arow + k);             // {A[m][k+khalf], A[m][k+khalf+1]}
        v2f bf;
        bf.x = wcol[(k + khalf)     * OUT_DIM] * msk; // hW[k+khalf  ][m]
        bf.y = wcol[(k + khalf + 1) * OUT_DIM] * msk; // hW[k+khalf+1][m]
        c = __builtin_amdgcn_wmma_f32_16x16x4_f32(
                /*neg_a=*/false, af, /*neg_b=*/false, bf,
                /*c_mod=*/(short)0, c, /*reuse_a=*/false, /*reuse_b=*/false);
    }

    if (valid) {
        const float hbn   = hb[m];
        const int   mbase = tile * 16 + ((lane < 16) ? 0 : 8);
#pragma unroll
        for (int r = 0; r < 8; ++r) {
            p[(size_t)(mbase + r) * OUT_DIM + m] = c[r] + hbn;
        }
    }
}

__global__ void binn_avg(float* __restrict__ avg, const float* __restrict__ stk, int n)
{
    int i = blockIdx.x * blockDim.x + threadIdx.x;
    if (i < n) avg[i] = (stk[i] + stk[i + n] + stk[i + 2 * n]) * (1.0f / 3.0f);
}

extern "C" void kernel_launch(void* const* d_in, const int* in_sizes, int n_in,
                              void* d_out, int out_size, void* d_ws, size_t ws_size,
                              hipStream_t stream)
{
    const float* x    = (const float*)d_in[0];
    const int*   src1 = (const int*)  d_in[1];
    const float* w1   = (const float*)d_in[2];
    const int*   src2 = (const int*)  d_in[3];
    const float* w2   = (const float*)d_in[4];
    const int*   src3 = (const int*)  d_in[5];
    const float* w3   = (const float*)d_in[6];
    // d_in[7]=src4, d_in[8]=w4: result discarded by the reference -> skipped
    const float* bias = (const float*)d_in[9];
    const float* g1v  = (const float*)d_in[10];
    const float* b1v  = (const float*)d_in[11];
    const float* g2v  = (const float*)d_in[12];
    const float* b2v  = (const float*)d_in[13];
    const float* g3v  = (const float*)d_in[14];
    const float* b3v  = (const float*)d_in[15];
    const float* hW1  = (const float*)d_in[16];
    const float* hb1  = (const float*)d_in[17];
    const float* hW2  = (const float*)d_in[18];
    const float* hb2  = (const float*)d_in[19];
    const float* hW3  = (const float*)d_in[20];
    const float* hb3  = (const float*)d_in[21];

    float* outp = (float*)d_out;
    float* avg  = outp;                              // [B, OUT]
    float* stk  = outp + BATCH * OUT_DIM;            // [3, B, OUT]

    // workspace: a1 (16MB) | a2 (8MB) | a3 (4MB) = 28MB
    float* a1 = (float*)d_ws;
    float* a2 = a1 + (size_t)BATCH * H1_DIM;
    float* a3 = a2 + (size_t)BATCH * H2_DIM;

    // ---- layer 1 ----
    binn_sparse<<<dim3(BATCH / 256, H1_DIM), 256, 0, stream>>>(
        x, src1, w1, bias, a1, IN_DIM, H1_DIM);
    binn_ln_tanh<<<BATCH, 256, 0, stream>>>(a1, g1v, b1v, H1_DIM);
    binn_head_wmma<<<BATCH / 16, 32, 0, stream>>>(
        a1, hW1, hb1, stk + 0 * BATCH * OUT_DIM, H1_DIM);

    // ---- layer 2 ----
    binn_sparse<<<dim3(BATCH / 256, H2_DIM), 256, 0, stream>>>(
        a1, src2, w2, bias + H1_DIM, a2, H1_DIM, H2_DIM);
    binn_ln_tanh<<<BATCH, 256, 0, stream>>>(a2, g2v, b2v, H2_DIM);
    binn_head_wmma<<<BATCH / 16, 32, 0, stream>>>(
        a2, hW2, hb2, stk + 1 * BATCH * OUT_DIM, H2_DIM);

    // ---- layer 3 ----
    binn_sparse<<<dim3(BATCH / 256, H3_DIM), 256, 0, stream>>>(
        a2, src3, w3, bias + H1_DIM + H2_DIM, a3, H2_DIM, H3_DIM);
    binn_ln_tanh<<<BATCH, 256, 0, stream>>>(a3, g3v, b3v, H3_DIM);
    binn_head_wmma<<<BATCH / 16, 32, 0, stream>>>(
        a3, hW3, hb3, stk + 2 * BATCH * OUT_DIM, H3_DIM);

    // ---- average of the three heads ----
    binn_avg<<<(BATCH * OUT_DIM + 255) / 256, 256, 0, stream>>>(
        avg, stk, BATCH * OUT_DIM);

    (void)in_sizes; (void)n_in; (void)out_size; (void)ws_size;
}